// SRWM_979252543657
// MI455X (gfx1250) — compile-verified
//
#include <hip/hip_runtime.h>
#include <math.h>

#define D_IN   1024
#define D_OUT  1024
#define BATCH  32
#define E_DIM  3076                      // D_OUT + 2*D_IN + 4
#define NTILE  ((E_DIM + 15) / 16)       // 193 row-tiles of 16

typedef __attribute__((ext_vector_type(2))) float v2f;
typedef __attribute__((ext_vector_type(4))) float v4f;
typedef __attribute__((ext_vector_type(8))) float v8f;

// ---------------------------------------------------------------------------
// Pass 1: out[b,e] = sum_d w[b,e,d] * x[b,d]     (V_WMMA_F32_16X16X4_F32)
// One wave = one 16-row tile. A = 16x4 chunk of W rows (ISA f32 A layout:
// lanes 0-15 hold k,k+1; lanes 16-31 hold k+2,k+3). B column 0 = x chunk.
// Dot products accumulate in D[:,0] (lane 0 -> rows 0-7, lane 16 -> rows 8-15).
// Tail rows are address-clamped only: their accumulators are garbage but are
// never stored (WMMA rows are independent), so no per-load masking needed.
// ---------------------------------------------------------------------------
__global__ __launch_bounds__(32)
void srwm_pass1(const float* __restrict__ w, const float* __restrict__ x,
                float* __restrict__ out_ws, float* __restrict__ y_out) {
  const int  lane = threadIdx.x & 31;
  const int  tile = blockIdx.x;
  const int  b    = blockIdx.y;
  const bool hi   = lane >= 16;
  const int  m    = lane & 15;
  const int  row  = tile * 16 + m;
  const int  rowc = (row < E_DIM) ? row : (E_DIM - 1);   // clamp, EXEC stays uniform

  const float* wrow = w + ((size_t)b * E_DIM + rowc) * D_IN + (hi ? 2 : 0);
  const float* xp   = x + (size_t)b * D_IN + (hi ? 2 : 0);

  v8f acc = {};
#pragma unroll 4
  for (int k = 0; k < D_IN; k += 4) {
    v2f a  = *(const v2f*)(wrow + k);
    v2f xv = *(const v2f*)(xp + k);
    v2f bb;
    bb.x = (m == 0) ? xv.x : 0.f;
    bb.y = (m == 0) ? xv.y : 0.f;
    acc = __builtin_amdgcn_wmma_f32_16x16x4_f32(false, a, false, bb,
                                                (short)0, acc, false, false);
  }

  if (m == 0) {
    const int mbase = hi ? 8 : 0;
#pragma unroll
    for (int j = 0; j < 8; ++j) {
      const int e = tile * 16 + mbase + j;
      if (e < E_DIM) {
        const float v = acc[j];
        out_ws[(size_t)b * E_DIM + e] = v;
        if (e < D_OUT) y_out[(size_t)b * D_OUT + e] = v;
      }
    }
  }
}

// ---------------------------------------------------------------------------
// Softmax over k (which=0) and q (which=1) segments, one block per (which,b).
// ---------------------------------------------------------------------------
__global__ __launch_bounds__(256)
void srwm_softmax(const float* __restrict__ out_ws,
                  float* __restrict__ kphi, float* __restrict__ qphi) {
  __shared__ float red[256];
  const int b     = blockIdx.y;
  const int which = blockIdx.x;
  const float* src = out_ws + (size_t)b * E_DIM + D_OUT + which * D_IN;
  float*       dst = (which ? qphi : kphi) + (size_t)b * D_IN;
  const int t = threadIdx.x;

  float vals[4];
  float mx = -INFINITY;
#pragma unroll
  for (int i = 0; i < 4; ++i) { vals[i] = src[t + i * 256]; mx = fmaxf(mx, vals[i]); }
  red[t] = mx; __syncthreads();
  for (int s = 128; s > 0; s >>= 1) {
    if (t < s) red[t] = fmaxf(red[t], red[t + s]);
    __syncthreads();
  }
  mx = red[0]; __syncthreads();

  float sum = 0.f;
#pragma unroll
  for (int i = 0; i < 4; ++i) { vals[i] = __expf(vals[i] - mx); sum += vals[i]; }
  red[t] = sum; __syncthreads();
  for (int s = 128; s > 0; s >>= 1) {
    if (t < s) red[t] += red[t + s];
    __syncthreads();
  }
  const float inv = 1.f / red[0];
#pragma unroll
  for (int i = 0; i < 4; ++i) dst[t + i * 256] = vals[i] * inv;
}

// ---------------------------------------------------------------------------
// Fused pass 2: per 16-row tile,
//   (a) vbar = W@kphi and v = W@qphi via two WMMAs sharing each A load,
//   (b) s[e]  = sigmoid(beta_e) * (v - vbar)   (lanes 0/16 -> LDS broadcast),
//   (c) w_out[e,:] = w[e,:] + s[e]*kphi[:]     (re-read of the tile hits L2).
// ---------------------------------------------------------------------------
__global__ __launch_bounds__(32)
void srwm_pass2_fused(const float* __restrict__ w,
                      const float* __restrict__ kphi,
                      const float* __restrict__ qphi,
                      const float* __restrict__ out_ws,
                      float* __restrict__ wout) {
  __shared__ float srow[16];
  const int  lane = threadIdx.x & 31;
  const int  tile = blockIdx.x;
  const int  b    = blockIdx.y;
  const bool hi   = lane >= 16;
  const int  m    = lane & 15;
  const int  row  = tile * 16 + m;
  const int  rowc = (row < E_DIM) ? row : (E_DIM - 1);

  const float* wrow = w    + ((size_t)b * E_DIM + rowc) * D_IN + (hi ? 2 : 0);
  const float* kp   = kphi + (size_t)b * D_IN + (hi ? 2 : 0);
  const float* qp   = qphi + (size_t)b * D_IN + (hi ? 2 : 0);

  v8f accK = {};
  v8f accQ = {};
#pragma unroll 2
  for (int k = 0; k < D_IN; k += 4) {
    v2f a  = *(const v2f*)(wrow + k);
    v2f kv = *(const v2f*)(kp + k);
    v2f qv = *(const v2f*)(qp + k);
    v2f bk, bq;
    bk.x = (m == 0) ? kv.x : 0.f;  bk.y = (m == 0) ? kv.y : 0.f;
    bq.x = (m == 0) ? qv.x : 0.f;  bq.y = (m == 0) ? qv.y : 0.f;
    accK = __builtin_amdgcn_wmma_f32_16x16x4_f32(false, a, false, bk,
                                                 (short)0, accK, false, false);
    accQ = __builtin_amdgcn_wmma_f32_16x16x4_f32(false, a, false, bq,
                                                 (short)0, accQ, false, false);
  }

  // Fold in sigmoid(beta) and publish per-row scale factors.
  if (m == 0) {
    const int    mbase = hi ? 8 : 0;
    const float* betas = out_ws + (size_t)b * E_DIM + (E_DIM - 4);
#pragma unroll
    for (int j = 0; j < 8; ++j) {
      const int e   = tile * 16 + mbase + j;        // tail rows: garbage, unused
      const int sel = (e < D_OUT) ? 0
                    : (e < D_OUT + D_IN) ? 1
                    : (e < D_OUT + 2 * D_IN) ? 2 : 3;
      const float beta = betas[sel];
      const float sg   = 1.f / (1.f + __expf(-beta));
      srow[mbase + j] = sg * (accQ[j] - accK[j]);
    }
  }
  __syncthreads();

  // Rank-1 update sweep: 32 lanes x float4 = 512B contiguous per instruction.
  const v4f* kp4 = (const v4f*)(kphi + (size_t)b * D_IN);
  for (int r = 0; r < 16; ++r) {
    const int e = tile * 16 + r;
    if (e >= E_DIM) break;                          // wave-uniform
    const float  s   = srow[r];
    const size_t off = ((size_t)b * E_DIM + e) * D_IN;
    const v4f* wp = (const v4f*)(w + off);
    v4f*       op = (v4f*)(wout + off);
#pragma unroll
    for (int i = 0; i < 8; ++i) {
      const int idx = lane + i * 32;
      op[idx] = wp[idx] + kp4[idx] * s;
    }
  }
}

// ---------------------------------------------------------------------------
extern "C" void kernel_launch(void* const* d_in, const int* in_sizes, int n_in,
                              void* d_out, int out_size, void* d_ws, size_t ws_size,
                              hipStream_t stream) {
  const float* x = (const float*)d_in[0];   // (32, 1024)
  const float* w = (const float*)d_in[1];   // (32, 3076, 1024)

  float* y    = (float*)d_out;                               // 32*1024
  float* wout = (float*)d_out + (size_t)BATCH * D_OUT;       // 32*3076*1024

  float* ws     = (float*)d_ws;
  float* out_ws = ws;                                        // 32*3076
  float* kphi   = out_ws + (size_t)BATCH * E_DIM;            // 32*1024
  float* qphi   = kphi   + (size_t)BATCH * D_IN;             // 32*1024

  srwm_pass1      <<<dim3(NTILE, BATCH), 32,  0, stream>>>(w, x, out_ws, y);
  srwm_softmax    <<<dim3(2,     BATCH), 256, 0, stream>>>(out_ws, kphi, qphi);
  srwm_pass2_fused<<<dim3(NTILE, BATCH), 32,  0, stream>>>(w, kphi, qphi, out_ws, wout);
}